// NeuralLogicMachine_81003083202897
// MI455X (gfx1250) — compile-verified
//
#include <hip/hip_runtime.h>

typedef __attribute__((ext_vector_type(2))) float v2f;
typedef __attribute__((ext_vector_type(8))) float v8f;

#define NOBJ    64
#define EMBD    64
#define NPRED   32
#define NLAY    3
#define BATCH   128
#define NPAIR   4032                    // 64*63
#define LBCNT   (NLAY*BATCH)            // 384
#define TILES_PER_LB (NPAIR/16)         // 252

static const size_t PREDS_ELEMS = (size_t)NLAY*BATCH*NPAIR*NPRED;  // 49,545,216
// workspace layout in floats
#define WS_MEAN 0                               // BATCH*EMBD        =   8192
#define WS_CB   (8192)                          // LBCNT*EMBD        =  24576
#define WS_AI   (8192 + 24576)                  // LBCNT*NOBJ*EMBD   = 1572864
#define WS_BJ   (WS_AI + 1572864)

// ---------------------------------------------------------------------------
// Kernel 1: embedding gather (also the second output) + per-batch mean
// grid = BATCH blocks, 64 threads (thread = feature d)
// ---------------------------------------------------------------------------
__global__ void k_gather_mean(const int* __restrict__ ids,
                              const float* __restrict__ emb,
                              float* __restrict__ e_out,
                              float* __restrict__ mean) {
  int b = blockIdx.x;
  int d = threadIdx.x;
  float s = 0.f;
  for (int n = 0; n < NOBJ; ++n) {
    int id = ids[b * NOBJ + n];
    float v = emb[id * EMBD + d];
    e_out[((size_t)b * NOBJ + n) * EMBD + d] = v;
    s += v;
  }
  mean[b * EMBD + d] = s * (1.0f / NOBJ);
}

// ---------------------------------------------------------------------------
// Kernel 2: per-object factorization of layer-1 GEMM.
//   Ai[lb,n,d] = e[b,n,:]·W1[l,d,0:64]
//   Bj[lb,n,d] = e[b,n,:]·W1[l,d,64:128]
//   Cb[lb,d]   = mean[b,:]·W1[l,d,128:192] + b1[l,d]
// grid = LBCNT blocks, 256 threads. E and W1a/W1b staged in LDS (W1 rows
// padded to 65 floats -> no 64-bank conflicts on the d-striped reads).
// ---------------------------------------------------------------------------
__global__ void k_precompute(const float* __restrict__ e,
                             const float* __restrict__ mean,
                             const float* __restrict__ W1,
                             const float* __restrict__ b1,
                             float* __restrict__ Ai,
                             float* __restrict__ Bj,
                             float* __restrict__ Cb) {
  int lb = blockIdx.x;
  int l  = lb / BATCH;
  int b  = lb - l * BATCH;
  int tid = threadIdx.x;

  __shared__ float sE[NOBJ * EMBD];     // 16 KB
  __shared__ float sWa[EMBD * 65];      // 16.25 KB (padded)
  __shared__ float sWb[EMBD * 65];      // 16.25 KB (padded)

  const float* W1l = W1 + (size_t)l * EMBD * (3 * EMBD);
  for (int o = tid; o < NOBJ * EMBD; o += 256) {
    int d = o >> 6, k = o & 63;
    sE[o]            = e[(size_t)b * NOBJ * EMBD + o];
    sWa[d * 65 + k]  = W1l[d * 192 + k];
    sWb[d * 65 + k]  = W1l[d * 192 + 64 + k];
  }
  __syncthreads();

  for (int o = tid; o < NOBJ * EMBD; o += 256) {
    int n = o >> 6, d = o & 63;
    float sa = 0.f, sb = 0.f;
#pragma unroll
    for (int k = 0; k < EMBD; ++k) {
      float ev = sE[n * EMBD + k];      // broadcast within 64-thread group
      sa += ev * sWa[d * 65 + k];
      sb += ev * sWb[d * 65 + k];
    }
    Ai[((size_t)lb * NOBJ + n) * EMBD + d] = sa;
    Bj[((size_t)lb * NOBJ + n) * EMBD + d] = sb;
  }

  if (tid < EMBD) {
    int d = tid;
    float s = b1[l * EMBD + d];
    for (int k = 0; k < EMBD; ++k)
      s += mean[b * EMBD + k] * W1l[d * 192 + 128 + k];
    Cb[lb * EMBD + d] = s;
  }
}

// ---------------------------------------------------------------------------
// Kernel 3: per-pair stage, fp32 WMMA (V_WMMA_F32_16X16X4_F32).
// One wave32 owns a tile of 16 pairs x 32 preds for one (l,b):
//   h(16x64) = relu(Ai[i] + Bj[j] + Cb)   built in registers in A-layout
//   D(16x32) = h @ W2[l]^T  via 16 K-chunks x 2 Q-tiles = 32 WMMAs
//   preds    = sigmoid(D + b2)
// grid = 96768 tiles / 8 waves = 12096 blocks x 256 threads, exact (no tail),
// EXEC all-ones as WMMA requires.
// ---------------------------------------------------------------------------
__global__ void k_pairs_wmma(const float* __restrict__ Ai,
                             const float* __restrict__ Bj,
                             const float* __restrict__ Cb,
                             const float* __restrict__ W2,
                             const float* __restrict__ b2,
                             float* __restrict__ preds) {
  int lane = threadIdx.x & 31;
  int wid  = threadIdx.x >> 5;
  int tile = blockIdx.x * 8 + wid;                 // 0 .. 96767
  int lb   = tile / TILES_PER_LB;
  int t16  = tile - lb * TILES_PER_LB;
  int l    = lb / BATCH;
  int p0   = t16 * 16;

  int m  = lane & 15;                              // pair-row within tile / Q col
  int hl = lane >> 4;                              // half-wave: K sub-split
  int p  = p0 + m;
  int i  = p / 63;
  int r  = p - i * 63;
  int j  = r + (r >= i ? 1 : 0);                   // ordered pairs, skip i==j

  const float* Arow = Ai + ((size_t)lb * NOBJ + i) * EMBD;
  const float* Brow = Bj + ((size_t)lb * NOBJ + j) * EMBD;
  const float* Crow = Cb + (size_t)lb * EMBD;

  // Build h in registers, already in 16x4 f32 A-operand layout:
  // lane m (lo half): K = 4t, 4t+1 ; lane m+16 (hi half): K = 4t+2, 4t+3
  v2f h[16];
#pragma unroll
  for (int t = 0; t < 16; ++t) {
    int d0 = 4 * t + 2 * hl;
    v2f av = *(const v2f*)(Arow + d0);
    v2f bv = *(const v2f*)(Brow + d0);
    v2f cv = *(const v2f*)(Crow + d0);
    float hx = av.x + bv.x + cv.x;
    float hy = av.y + bv.y + cv.y;
    h[t].x = hx > 0.f ? hx : 0.f;
    h[t].y = hy > 0.f ? hy : 0.f;
  }

  // B operand: W2^T tiles. B[k][n] = W2[l][q0+n][k]; lane supplies column m,
  // K rows (4t+2*hl, 4t+2*hl+1) -> contiguous float2 from W2 row q0+m.
  const float* w2q0 = W2 + ((size_t)l * NPRED + m) * EMBD;   // q-tile 0
  const float* w2q1 = w2q0 + 16 * EMBD;                      // q-tile 1

  v8f acc0 = {};
  v8f acc1 = {};
#pragma unroll
  for (int t = 0; t < 16; ++t) {
    int d0 = 4 * t + 2 * hl;
    v2f b0 = *(const v2f*)(w2q0 + d0);
    v2f b1v = *(const v2f*)(w2q1 + d0);
    acc0 = __builtin_amdgcn_wmma_f32_16x16x4_f32(false, h[t], false, b0,
                                                 (short)0, acc0, false, false);
    acc1 = __builtin_amdgcn_wmma_f32_16x16x4_f32(false, h[t], false, b1v,
                                                 (short)0, acc1, false, false);
  }

  // Epilogue: D layout -> VGPR v holds row v (lanes 0-15) / row v+8 (16-31),
  // column = q0 + (lane&15). Add bias, sigmoid, store.
  float bias0 = b2[l * NPRED + m];
  float bias1 = b2[l * NPRED + 16 + m];
  size_t outbase = ((size_t)lb * NPAIR + p0) * NPRED;
#pragma unroll
  for (int v = 0; v < 8; ++v) {
    int row = v + 8 * hl;
    float x0 = acc0[v] + bias0;
    float x1 = acc1[v] + bias1;
    x0 = 1.f / (1.f + __expf(-x0));
    x1 = 1.f / (1.f + __expf(-x1));
    preds[outbase + (size_t)row * NPRED + m]      = x0;
    preds[outbase + (size_t)row * NPRED + 16 + m] = x1;
  }
}

// ---------------------------------------------------------------------------
extern "C" void kernel_launch(void* const* d_in, const int* in_sizes, int n_in,
                              void* d_out, int out_size, void* d_ws, size_t ws_size,
                              hipStream_t stream) {
  const int*   ids = (const int*)  d_in[0];
  const float* emb = (const float*)d_in[1];
  const float* W1  = (const float*)d_in[2];
  const float* b1  = (const float*)d_in[3];
  const float* W2  = (const float*)d_in[4];
  const float* b2  = (const float*)d_in[5];

  float* out   = (float*)d_out;
  float* preds = out;                    // [L,B,P,Q]
  float* e_out = out + PREDS_ELEMS;      // [B,N,D] second output

  float* ws   = (float*)d_ws;
  float* mean = ws + WS_MEAN;
  float* Cb   = ws + WS_CB;
  float* Ai   = ws + WS_AI;
  float* Bj   = ws + WS_BJ;

  k_gather_mean<<<BATCH, 64, 0, stream>>>(ids, emb, e_out, mean);
  k_precompute<<<LBCNT, 256, 0, stream>>>(e_out, mean, W1, b1, Ai, Bj, Cb);
  k_pairs_wmma<<<(LBCNT * TILES_PER_LB) / 8, 256, 0, stream>>>(Ai, Bj, Cb, W2, b2, preds);
}